// GATv2TrajectoryPredictor_14362370638061
// MI455X (gfx1250) — compile-verified
//
#include <hip/hip_runtime.h>

// ---------------------------------------------------------------------------
// GATv2 trajectory predictor for gfx1250 (MI455X).
// Dense projections: V_WMMA_F32_16X16X4_F32, one wave = 16x64 output strip
// (A fragment loaded as one b64 and reused across 4 WMMAs per K-step).
// Edge softmax/aggregation: monotone-uint atomicMax + f32 global atomics.
// ---------------------------------------------------------------------------

#define HIDC 64
#define NHEADS 4

typedef float v2f __attribute__((ext_vector_type(2)));
typedef float v8f __attribute__((ext_vector_type(8)));

// monotone float <-> uint encoding for atomicMax-based segment max
__device__ __forceinline__ unsigned fenc(float f) {
  unsigned u = __float_as_uint(f);
  return (u & 0x80000000u) ? ~u : (u | 0x80000000u);
}
__device__ __forceinline__ float fdec(unsigned u) {
  return __uint_as_float((u & 0x80000000u) ? (u & 0x7fffffffu) : ~u);
}

// ---------------------------------------------------------------------------
// f32 GEMM: out[M,Nc] = act(A[M,K] @ W[K,Nc] + bias)
// Requirements: M % 16 == 0, Nc % 64 == 0, K % 4 == 0.
// One wave computes a 16x64 strip = 4 16x16 tiles via V_WMMA_F32_16X16X4_F32.
// A frag (16x4): lanes 0-15 hold K=kk,kk+1; lanes 16-31 hold K=kk+2,kk+3
//   -> one contiguous float2 (global_load_b64) per lane per K-step.
// B frag (4x16): same K split, N = lane&15.
// C/D: vgpr r -> row (r, or r+8 for lanes 16-31), col = lane&15.
// act: 0=none, 1=relu
// ---------------------------------------------------------------------------
__global__ __launch_bounds__(256) void k_gemm_wmma(
    const float* __restrict__ A, const float* __restrict__ W,
    const float* __restrict__ bias, float* __restrict__ out,
    int M, int Nc, int K, int act)
{
  const int lane = threadIdx.x & 31;
  const int wave = threadIdx.x >> 5;
  const int stripsN = Nc >> 6;                       // strips of 4 N-tiles
  const long nStrips = (long)(M >> 4) * stripsN;
  const long strip = (long)blockIdx.x * 8 + wave;
  if (strip >= nStrips) return;      // uniform per wave: EXEC stays all-1s

  const int tm = (int)(strip / stripsN);
  const int tn = (int)(strip % stripsN);             // strip index (64 cols)
  const int half = lane >> 4;                        // 0: K {0,1}, 1: K {2,3}
  const int l15 = lane & 15;
  const long rowA = (long)tm * 16 + l15;
  const int col0 = tn * 64 + l15;

  v8f acc0 = {}, acc1 = {}, acc2 = {}, acc3 = {};
  const float* arow = A + rowA * K + half * 2;

  for (int kk = 0; kk < K; kk += 4) {
    const float2 af = *(const float2*)(arow + kk);   // contiguous K pair
    v2f a; a.x = af.x; a.y = af.y;
    const float* wr0 = W + (long)(kk + half * 2) * Nc + col0; // row k0
    const float* wr1 = wr0 + Nc;                              // row k0+1
    v2f b0, b1, b2, b3;                 // load all B frags, then 4 WMMAs
    b0.x = wr0[0];  b0.y = wr1[0];
    b1.x = wr0[16]; b1.y = wr1[16];
    b2.x = wr0[32]; b2.y = wr1[32];
    b3.x = wr0[48]; b3.y = wr1[48];
    acc0 = __builtin_amdgcn_wmma_f32_16x16x4_f32(false, a, false, b0, (short)0, acc0, false, false);
    acc1 = __builtin_amdgcn_wmma_f32_16x16x4_f32(false, a, false, b1, (short)0, acc1, false, false);
    acc2 = __builtin_amdgcn_wmma_f32_16x16x4_f32(false, a, false, b2, (short)0, acc2, false, false);
    acc3 = __builtin_amdgcn_wmma_f32_16x16x4_f32(false, a, false, b3, (short)0, acc3, false, false);
  }

  const int rowBase = tm * 16 + half * 8;
  v8f accs[4] = {acc0, acc1, acc2, acc3};
#pragma unroll
  for (int j = 0; j < 4; ++j) {
    const int col = col0 + j * 16;
    const float bv = bias ? bias[col] : 0.0f;
#pragma unroll
    for (int r = 0; r < 8; ++r) {
      float v = accs[j][r] + bv;
      if (act == 1) v = fmaxf(v, 0.0f);
      out[(long)(rowBase + r) * Nc + col] = v;
    }
  }
}

// zero-pad x[N,10] -> xp[N,12] so the embedding GEMM needs no K guards
__global__ void k_pad_embed(const float* __restrict__ x, float* __restrict__ xp,
                            long total /* = N*12 */)
{
  long t = (long)blockIdx.x * blockDim.x + threadIdx.x;
  if (t >= total) return;
  const int k = (int)(t % 12);
  const long n = t / 12;
  xp[t] = (k < 10) ? x[n * 10 + k] : 0.0f;
}

// zero-pad We[10,64] -> Wp[12,64] (rows 10,11 = 0) so GEMM reads stay in-bounds
__global__ void k_pad_we(const float* __restrict__ We, float* __restrict__ Wp) {
  const int t = blockIdx.x * blockDim.x + threadIdx.x;   // 0 .. 12*64-1
  if (t >= 12 * HIDC) return;
  Wp[t] = (t < 10 * HIDC) ? We[t] : 0.0f;
}

// zero-fill (also initializes encoded segment-max: 0u decodes below any real)
__global__ void k_zero(unsigned* __restrict__ p, long n) {
  long t = (long)blockIdx.x * blockDim.x + threadIdx.x;
  if (t < n) p[t] = 0u;
}

// ---------------------------------------------------------------------------
// Pass 1: per (edge, head) attention logit + segment max via atomicMax.
// Edges e<E come from edge_index; e>=E are self loops.
// ---------------------------------------------------------------------------
__global__ void k_edge_logits(
    const float* __restrict__ xl, const float* __restrict__ xr,
    const int* __restrict__ ei, const float* __restrict__ att,
    float* __restrict__ logits, unsigned* __restrict__ mmax,
    int Ecnt, int Nn, int heads, int C)
{
  long t = (long)blockIdx.x * blockDim.x + threadIdx.x;
  const long total = (long)(Ecnt + Nn) * heads;
  if (t >= total) return;
  const int h = (int)(t % heads);
  const long e = t / heads;
  int s, d;
  if (e < Ecnt) { s = ei[e]; d = ei[Ecnt + e]; }
  else          { s = d = (int)(e - Ecnt); }

  const float4* pl = (const float4*)(xl + ((long)s * heads + h) * C);
  const float4* pr = (const float4*)(xr + ((long)d * heads + h) * C);
  const float4* pa = (const float4*)(att + (long)h * C);
  float acc = 0.0f;
  const int c4n = C >> 2;
  for (int i = 0; i < c4n; ++i) {
    float4 a = pl[i], b = pr[i], w = pa[i];
    float v;
    v = a.x + b.x; v = (v > 0.0f) ? v : 0.2f * v; acc += v * w.x;
    v = a.y + b.y; v = (v > 0.0f) ? v : 0.2f * v; acc += v * w.y;
    v = a.z + b.z; v = (v > 0.0f) ? v : 0.2f * v; acc += v * w.z;
    v = a.w + b.w; v = (v > 0.0f) ? v : 0.2f * v; acc += v * w.w;
  }
  logits[t] = acc;
  atomicMax(mmax + (long)d * heads + h, fenc(acc));
}

// ---------------------------------------------------------------------------
// Pass 2: ex = exp(logit - max[dst]); den[dst] += ex; overwrite logits with ex
// ---------------------------------------------------------------------------
__global__ void k_edge_den(
    float* __restrict__ logits, const unsigned* __restrict__ mmax,
    const int* __restrict__ ei, float* __restrict__ den,
    int Ecnt, int Nn, int heads)
{
  long t = (long)blockIdx.x * blockDim.x + threadIdx.x;
  const long total = (long)(Ecnt + Nn) * heads;
  if (t >= total) return;
  const int h = (int)(t % heads);
  const long e = t / heads;
  const int d = (e < Ecnt) ? ei[Ecnt + e] : (int)(e - Ecnt);
  const float m = fdec(mmax[(long)d * heads + h]);
  const float ex = __expf(logits[t] - m);
  logits[t] = ex;
  atomicAdd(den + (long)d * heads + h, ex);
}

// ---------------------------------------------------------------------------
// Pass 3: numerator scatter. Thread per (edge, head, 4-channel group):
// acc[dst,h,c] += xl[src,h,c] * ex.  Consecutive threads hit consecutive
// channels -> coalesced global_atomic_add_f32.
// ---------------------------------------------------------------------------
__global__ void k_edge_num(
    const float* __restrict__ xl, const int* __restrict__ ei,
    const float* __restrict__ ex, float* __restrict__ acc,
    int Ecnt, int Nn, int heads, int C)
{
  long t = (long)blockIdx.x * blockDim.x + threadIdx.x;
  const int c4n = C >> 2;
  const long total = (long)(Ecnt + Nn) * heads * c4n;
  if (t >= total) return;
  const int c4 = (int)(t % c4n);
  const long eh = t / c4n;
  const int h = (int)(eh % heads);
  const long e = eh / heads;
  int s, d;
  if (e < Ecnt) { s = ei[e]; d = ei[Ecnt + e]; }
  else          { s = d = (int)(e - Ecnt); }
  const float w = ex[e * heads + h];
  const float4 v = ((const float4*)(xl + ((long)s * heads + h) * C))[c4];
  float* dst = acc + ((long)d * heads + h) * C + c4 * 4;
  atomicAdd(dst + 0, v.x * w);
  atomicAdd(dst + 1, v.y * w);
  atomicAdd(dst + 2, v.z * w);
  atomicAdd(dst + 3, v.w * w);
}

// h = act(acc/den + bias) in place. act: 0=none, 1=elu(alpha=1)
__global__ void k_finalize(
    float* __restrict__ hacc, const float* __restrict__ den,
    const float* __restrict__ bias, long total, int heads, int C, int act)
{
  long t = (long)blockIdx.x * blockDim.x + threadIdx.x;
  if (t >= total) return;
  const int c = (int)(t % C);
  const long nh = t / C;
  const int h = (int)(nh % heads);
  float v = hacc[t] / den[nh] + bias[h * C + c];
  if (act) v = (v > 0.0f) ? v : (__expf(v) - 1.0f);
  hacc[t] = v;
}

// Final head: out[f,o] = concat(h3[focal[f]], mean(cl[f])@Wc+bc) @ Wf + bf
__global__ void k_head(
    const float* __restrict__ h3, const int* __restrict__ focal,
    const float* __restrict__ cl, const float* __restrict__ Wc,
    const float* __restrict__ bc, const float* __restrict__ Wf,
    const float* __restrict__ bf, float* __restrict__ out, int L)
{
  const int f = blockIdx.x;
  const int o = threadIdx.x;
  if (o >= 60) return;
  const int node = focal[f];
  float mx = 0.0f, my = 0.0f;
  for (int j = 0; j < L; ++j) {
    mx += cl[((long)f * L + j) * 2 + 0];
    my += cl[((long)f * L + j) * 2 + 1];
  }
  const float inv = 1.0f / (float)L;
  mx *= inv; my *= inv;
  float acc = bf[o];
  for (int k = 0; k < HIDC; ++k)
    acc += h3[(long)node * HIDC + k] * Wf[k * 60 + o];
  for (int k = 0; k < HIDC; ++k) {
    const float cfeat = mx * Wc[k] + my * Wc[HIDC + k] + bc[k];
    acc += cfeat * Wf[(HIDC + k) * 60 + o];
  }
  out[f * 60 + o] = acc;
}

static inline unsigned nblk(long n) { return (unsigned)((n + 255) / 256); }

extern "C" void kernel_launch(void* const* d_in, const int* in_sizes, int n_in,
                              void* d_out, int out_size, void* d_ws, size_t ws_size,
                              hipStream_t stream) {
  const float* x    = (const float*)d_in[0];
  const int*   ei   = (const int*)  d_in[1];
  const int*   focal= (const int*)  d_in[2];
  const float* cl   = (const float*)d_in[3];
  const float* We   = (const float*)d_in[4];  const float* be   = (const float*)d_in[5];
  const float* Wl1  = (const float*)d_in[6];  const float* bl1  = (const float*)d_in[7];
  const float* Wr1  = (const float*)d_in[8];  const float* br1  = (const float*)d_in[9];
  const float* att1 = (const float*)d_in[10]; const float* bias1= (const float*)d_in[11];
  const float* Wl2  = (const float*)d_in[12]; const float* bl2  = (const float*)d_in[13];
  const float* Wr2  = (const float*)d_in[14]; const float* br2  = (const float*)d_in[15];
  const float* att2 = (const float*)d_in[16]; const float* bias2= (const float*)d_in[17];
  const float* Wl3  = (const float*)d_in[18]; const float* bl3  = (const float*)d_in[19];
  const float* Wr3  = (const float*)d_in[20]; const float* br3  = (const float*)d_in[21];
  const float* att3 = (const float*)d_in[22]; const float* bias3= (const float*)d_in[23];
  const float* Wc   = (const float*)d_in[24]; const float* bc   = (const float*)d_in[25];
  const float* Wf   = (const float*)d_in[26]; const float* bf   = (const float*)d_in[27];

  const int N  = in_sizes[0] / 10;   // 50000
  const int E  = in_sizes[1] / 2;    // 800000
  const int NF = in_sizes[2];        // 64
  const int L  = in_sizes[3] / (NF * 2); // 50
  const long ET = (long)E + N;       // edges + self loops

  // workspace layout (floats)
  const size_t NR = (size_t)N * (NHEADS * HIDC);    // N*256
  float*    R0 = (float*)d_ws;           // h / attention accumulator (ping-pong)
  float*    R1 = R0 + NR;                // xl
  float*    R2 = R1 + NR;                // xr
  unsigned* Rm = (unsigned*)(R2 + NR);   // segment max, N*heads
  float*    Rd = (float*)(Rm + (size_t)N * NHEADS);  // denominator, N*heads
  float*    Rl = Rd + (size_t)N * NHEADS;            // logits/ex, ET*heads
  float*    Xp = Rl + (size_t)ET * NHEADS;           // padded input, N*12
  float*    Wp = Xp + (size_t)N * 12;                // padded embed weight, 12*64

  // ---- agent embedding: h0 = relu(Xp[N,12] @ Wp[12,64] + be) -> R0 ----
  k_pad_embed<<<nblk((long)N * 12), 256, 0, stream>>>(x, Xp, (long)N * 12);
  k_pad_we<<<3, 256, 0, stream>>>(We, Wp);
  {
    const long strips = (long)(N / 16) * (HIDC / 64);
    k_gemm_wmma<<<(unsigned)((strips + 7) / 8), 256, 0, stream>>>(
        Xp, Wp, be, R0, N, HIDC, 12, /*act=*/1);
  }

  // ---- one GATv2 layer ----
  auto layer = [&](const float* Wl, const float* bl, const float* Wr,
                   const float* br, const float* att, const float* bias,
                   int Kin, int heads, int act) {
    const int NC = heads * HIDC;
    const long strips = (long)(N / 16) * (NC / 64);
    // projections (WMMA GEMMs)
    k_gemm_wmma<<<(unsigned)((strips + 7) / 8), 256, 0, stream>>>(
        R0, Wl, bl, R1, N, NC, Kin, 0);
    k_gemm_wmma<<<(unsigned)((strips + 7) / 8), 256, 0, stream>>>(
        R0, Wr, br, R2, N, NC, Kin, 0);
    // init accumulator (reuses h region), segment max, denominator
    k_zero<<<nblk((long)N * NC), 256, 0, stream>>>((unsigned*)R0, (long)N * NC);
    k_zero<<<nblk((long)N * heads), 256, 0, stream>>>(Rm, (long)N * heads);
    k_zero<<<nblk((long)N * heads), 256, 0, stream>>>((unsigned*)Rd, (long)N * heads);
    // edge phase
    k_edge_logits<<<nblk(ET * heads), 256, 0, stream>>>(
        R1, R2, ei, att, Rl, Rm, E, N, heads, HIDC);
    k_edge_den<<<nblk(ET * heads), 256, 0, stream>>>(
        Rl, Rm, ei, Rd, E, N, heads);
    k_edge_num<<<nblk(ET * heads * (HIDC / 4)), 256, 0, stream>>>(
        R1, ei, Rl, R0, E, N, heads, HIDC);
    // normalize + bias + activation (in place -> next layer's h)
    k_finalize<<<nblk((long)N * NC), 256, 0, stream>>>(
        R0, Rd, bias, (long)N * NC, heads, HIDC, act);
  };

  layer(Wl1, bl1, Wr1, br1, att1, bias1, /*Kin=*/HIDC,          NHEADS, /*elu=*/1);
  layer(Wl2, bl2, Wr2, br2, att2, bias2, /*Kin=*/NHEADS * HIDC, NHEADS, /*elu=*/1);
  layer(Wl3, bl3, Wr3, br3, att3, bias3, /*Kin=*/NHEADS * HIDC, 1,      /*elu=*/0);

  // ---- prediction head ----
  k_head<<<NF, 64, 0, stream>>>(R0, focal, cl, Wc, bc, Wf, bf, (float*)d_out, L);
}